// DiagScanModule_2860448219143
// MI455X (gfx1250) — compile-verified
//
#include <hip/hip_runtime.h>
#include <cstdint>
#include <cstddef>

// DiagScan permutation for MI455X (gfx1250, wave32).
//
// Pure permutation kernel: min traffic = 512MB read (x once) + 1GB write
// => ~64us floor at 23.3 TB/s HBM. No FLOPs -> WMMA not applicable; the
// CDNA5 paths exercised are global_load_async_to_lds_b128 (+ ASYNCcnt wait)
// for index staging and NT-hinted b128 stores for the streaming outputs.

typedef float v4f __attribute__((ext_vector_type(4)));
typedef int   v4i __attribute__((ext_vector_type(4)));

#define THREADS 256
#define VPT     4                 // elements per thread per map
#define CHUNK   (THREADS * VPT)   // 1024 output positions per block
#define HH      512
#define WW      512
#define HWSZ    (HH * WW)         // 262144, divisible by CHUNK (256 chunks)
#define NSLICE  (8 * 64)          // B*C = 512

__global__ __launch_bounds__(THREADS)
void diag_gather_kernel(const float* __restrict__ x,
                        const int*  __restrict__ rd_map,
                        const int*  __restrict__ ld_map,
                        float* __restrict__ out)
{
    __shared__ alignas(16) int s_rd[CHUNK];
    __shared__ alignas(16) int s_ld[CHUNK];

    const int tid   = threadIdx.x;
    const int slice = blockIdx.y;                       // 0..511 : (b,c) slice
    const int pbase = blockIdx.x * CHUNK + tid * VPT;   // output pos of 4 elems

    // ---- gfx1250 async global->LDS staging of the two 4KB index chunks ----
    // Each lane async-copies its own 16B slot and reads back only that slot,
    // so s_wait_asynccnt 0 (per-wave) is sufficient ordering; no barrier.
    {
        uint64_t g0 = (uint64_t)(uintptr_t)(rd_map + pbase);
        uint64_t g1 = (uint64_t)(uintptr_t)(ld_map + pbase);
        uint32_t l0 = (uint32_t)(uintptr_t)(&s_rd[tid * VPT]);
        uint32_t l1 = (uint32_t)(uintptr_t)(&s_ld[tid * VPT]);
        asm volatile(
            "global_load_async_to_lds_b128 %0, %1, off\n\t"
            "global_load_async_to_lds_b128 %2, %3, off"
            :
            : "v"(l0), "v"(g0), "v"(l1), "v"(g1)
            : "memory");
        asm volatile("s_wait_asynccnt 0" ::: "memory");
    }

    const v4i ir = *(const v4i*)&s_rd[tid * VPT];   // ds_load_b128
    const v4i il = *(const v4i*)&s_ld[tid * VPT];

    // Scattered reads: stride (W-1) within a diagonal, but adjacent diagonals
    // reuse the same 128B lines (32x reuse) -> slice stays hot in L2, HBM
    // reads x exactly once. Keep these cached (regular temporal loads).
    const float* __restrict__ xs = x + (size_t)slice * HWSZ;

    v4f a, b;
    a.x = xs[ir.x]; a.y = xs[ir.y]; a.z = xs[ir.z]; a.w = xs[ir.w];
    b.x = xs[il.x]; b.y = xs[il.y]; b.z = xs[il.z]; b.w = xs[il.w];

    // Streaming outputs: coalesced b128 stores with non-temporal hint so the
    // 1GB write stream does not evict the reused x lines / index maps in L2.
    float* yrd = out + (size_t)slice * HWSZ + pbase;
    float* yld = out + (size_t)(NSLICE + slice) * HWSZ + pbase;
    __builtin_nontemporal_store(a, (v4f*)yrd);
    __builtin_nontemporal_store(b, (v4f*)yld);
}

extern "C" void kernel_launch(void* const* d_in, const int* in_sizes, int n_in,
                              void* d_out, int out_size, void* d_ws, size_t ws_size,
                              hipStream_t stream)
{
    (void)in_sizes; (void)n_in; (void)out_size; (void)d_ws; (void)ws_size;

    const float* x  = (const float*)d_in[0];
    const int*   rd = (const int*)d_in[1];
    const int*   ld = (const int*)d_in[2];
    float*       out = (float*)d_out;

    dim3 grid(HWSZ / CHUNK, NSLICE);   // (256, 512)
    diag_gather_kernel<<<grid, THREADS, 0, stream>>>(x, rd, ld, out);
}